// WordAttention_34522947125977
// MI455X (gfx1250) — compile-verified
//
#include <hip/hip_runtime.h>
#include <hip/hip_bf16.h>
#include <math.h>

#define HIDDEN 128

typedef __attribute__((ext_vector_type(2))) float v2f;
typedef __attribute__((ext_vector_type(8))) float v8f;

// ---------------------------------------------------------------- init ----
__global__ void zero_out_kernel(float4* __restrict__ out, int n4) {
    int i = blockIdx.x * blockDim.x + threadIdx.x;
    int stride = gridDim.x * blockDim.x;
    for (; i < n4; i += stride) out[i] = make_float4(0.f, 0.f, 0.f, 0.f);
}

__global__ void init_hdr_kernel(float* __restrict__ hdr) {
    if (threadIdx.x == 0) { hdr[0] = 0.0f; hdr[1] = 0.0f; }
}

// -------------------------------------------------- stage 1: WMMA scores --
// One wave32 per 16 nodes. D = A(16x4 f32) * B(4x16 f32) + C accumulated over
// K=128 in steps of 4 -> 32 v_wmma_f32_16x16x4_f32 per wave.
// A layout (ISA 7.12.2, 32-bit A 16x4): lanes 0-15 hold M=lane with
//   VGPR0=K0, VGPR1=K1; lanes 16-31 hold M=lane-16 with VGPR0=K2, VGPR1=K3.
// B (4x16) mirrors that K split; all 16 columns carry the same W value so
// every output column equals the dot product -> read column N=0.
__global__ __launch_bounds__(32) void scores_kernel(
        const float* __restrict__ x, const float* __restrict__ Watt,
        const float* __restrict__ batt, float* __restrict__ scores,
        int n_nodes) {
    const int lane = threadIdx.x;          // 0..31, EXEC all ones
    const int m    = lane & 15;
    const int koff = (lane < 16) ? 0 : 2;
    int nodeIdx = blockIdx.x * 16 + m;
    int srcIdx  = (nodeIdx < n_nodes) ? nodeIdx : (n_nodes - 1); // clamp, no divergence
    const float* __restrict__ xrow = x + (size_t)srcIdx * HIDDEN;

    v8f c = {0.f, 0.f, 0.f, 0.f, 0.f, 0.f, 0.f, 0.f};
#pragma unroll
    for (int kb = 0; kb < HIDDEN; kb += 4) {
        v2f a, b;
        a.x = xrow[kb + koff];
        a.y = xrow[kb + koff + 1];
        b.x = Watt[kb + koff];
        b.y = Watt[kb + koff + 1];
        // 8 args: (neg_a, A, neg_b, B, c_mod, C, reuse_a, reuse_b)
        c = __builtin_amdgcn_wmma_f32_16x16x4_f32(
                false, a, false, b, (short)0, c, false, false);
    }

    const float bias = batt[0];
    // Column N=0 of D lives in lane 0 (M=0..7 in c[0..7]) and lane 16 (M=8..15).
    if (lane == 0 || lane == 16) {
        int mbase = blockIdx.x * 16 + ((lane == 16) ? 8 : 0);
#pragma unroll
        for (int j = 0; j < 8; ++j) {
            int nd = mbase + j;
            if (nd < n_nodes) {
                float z = c[j] + bias;
                scores[nd] = 1.0f / (1.0f + __expf(-z));
            }
        }
    }
}

// ---------------------------------------- stage 2: edge energy + global max
__global__ __launch_bounds__(256) void energy_kernel(
        const int* __restrict__ row, const int* __restrict__ col,
        const float* __restrict__ wgt, const float* __restrict__ scores,
        float* __restrict__ energy, float* __restrict__ hdr, int E) {
    __shared__ float red[256];
    int t = blockIdx.x * 256 + threadIdx.x;
    float e = 0.0f;                         // identity for max of nonneg values
    if (t < E) {
        e = scores[row[t]] * scores[col[t]] * wgt[t];
        energy[t] = e;
    }
    red[threadIdx.x] = e;
    __syncthreads();
#pragma unroll
    for (int s = 128; s > 0; s >>= 1) {
        if (threadIdx.x < s)
            red[threadIdx.x] = fmaxf(red[threadIdx.x], red[threadIdx.x + s]);
        __syncthreads();
    }
    // energies >= 0  ->  IEEE float ordering == int ordering on the bits
    if (threadIdx.x == 0) atomicMax((int*)hdr, __float_as_int(red[0]));
}

// ------------------------------------------- stage 3: exp(e - max) + sum --
__global__ __launch_bounds__(256) void expsum_kernel(
        float* __restrict__ energy, float* __restrict__ hdr, int E) {
    __shared__ float red[256];
    const float mx = hdr[0];
    int t = blockIdx.x * 256 + threadIdx.x;
    float ex = 0.0f;
    if (t < E) {
        ex = __expf(energy[t] - mx);
        energy[t] = ex;                     // overwrite in place with exp value
    }
    red[threadIdx.x] = ex;
    __syncthreads();
#pragma unroll
    for (int s = 128; s > 0; s >>= 1) {
        if (threadIdx.x < s) red[threadIdx.x] += red[threadIdx.x + s];
        __syncthreads();
    }
    if (threadIdx.x == 0) atomicAdd(&hdr[1], red[0]);
}

// -------------------------------- stage 4: gather x[col], scatter to row --
// One wave32 per edge; lane handles a float4 (128 floats / 32 lanes).
__global__ __launch_bounds__(256) void scatter_kernel(
        const float4* __restrict__ x4, const int* __restrict__ row,
        const int* __restrict__ col, const float* __restrict__ ex,
        const float* __restrict__ hdr, float* __restrict__ out, int E) {
    const int wave = threadIdx.x >> 5;
    const int lane = threadIdx.x & 31;
    int e = blockIdx.x * 8 + wave;
    if (e >= E) return;
    const float coef = ex[e] * (1.0f / hdr[1]);
    const int r = row[e];
    const int c = col[e];
    float4 xv = x4[(size_t)c * (HIDDEN / 4) + lane];
    float* __restrict__ op = out + (size_t)r * HIDDEN + lane * 4;
    atomicAdd(op + 0, xv.x * coef);  // no-return global_atomic_add_f32
    atomicAdd(op + 1, xv.y * coef);
    atomicAdd(op + 2, xv.z * coef);
    atomicAdd(op + 3, xv.w * coef);
}

// ----------------------------------------------------------------- launch --
extern "C" void kernel_launch(void* const* d_in, const int* in_sizes, int n_in,
                              void* d_out, int out_size, void* d_ws, size_t ws_size,
                              hipStream_t stream) {
    const float* x    = (const float*)d_in[0];
    const int*   eidx = (const int*)d_in[1];
    const float* wgt  = (const float*)d_in[2];
    const float* Watt = (const float*)d_in[3];
    const float* batt = (const float*)d_in[4];

    const int E = in_sizes[2];              // 1,600,000
    const int N = in_sizes[0] / HIDDEN;     // 100,000
    const int* row = eidx;                  // edge_index[0]
    const int* col = eidx + E;              // edge_index[1]
    float* out = (float*)d_out;

    // ws layout: [0..255] header (hdr[0]=max, hdr[1]=sum),
    //            then scores[N], then energy/exp[E]
    char*  ws      = (char*)d_ws;
    float* hdr     = (float*)ws;
    float* scores  = (float*)(ws + 256);
    size_t eoff    = 256 + ((((size_t)N * sizeof(float)) + 255) & ~(size_t)255);
    float* energy  = (float*)(ws + eoff);

    zero_out_kernel<<<4096, 256, 0, stream>>>((float4*)out, out_size / 4);
    init_hdr_kernel<<<1, 32, 0, stream>>>(hdr);

    scores_kernel<<<(N + 15) / 16, 32, 0, stream>>>(x, Watt, batt, scores, N);

    int nb = (E + 255) / 256;
    energy_kernel<<<nb, 256, 0, stream>>>(row, col, wgt, scores, energy, hdr, E);
    expsum_kernel<<<nb, 256, 0, stream>>>(energy, hdr, E);

    scatter_kernel<<<(E + 7) / 8, 256, 0, stream>>>(
        (const float4*)x, row, col, energy, hdr, out, E);
}